// Registration_8375186227745
// MI455X (gfx1250) — compile-verified
//
#include <hip/hip_runtime.h>
#include <hip/hip_bf16.h>
#include <math.h>

#define VD 256           // volume dimension (256^3, from in_sizes[0])
#define TILE 16          // block covers TILE x TILE pixels

typedef __attribute__((ext_vector_type(2))) float v2f;
typedef __attribute__((ext_vector_type(8))) float v8f;

// ---------------------------------------------------------------------------
// Kernel 1: pose math on a single thread.
// Computes ext = rt @ (pivot @ T @ pivot_inv), then stores
//   ws[0..8]  = M3 = sdd * ext[:3,:3] @ k_inv   (row major)
//   ws[9..11] = src = ext[:3,3]
// ---------------------------------------------------------------------------
__global__ void reg_setup_kernel(const float* __restrict__ iso,
                                 const float* __restrict__ rt,     // 4x4
                                 const float* __restrict__ kinv,   // 3x3
                                 const float* __restrict__ sddp,   // 1
                                 const float* __restrict__ rot,    // 3
                                 const float* __restrict__ xyz,    // 3
                                 float* __restrict__ ws) {
    if (threadIdx.x != 0 || blockIdx.x != 0) return;

    // ---- 4x4 inverse of rt via Gauss-Jordan with partial pivoting ----
    float a[4][8];
    for (int i = 0; i < 4; ++i)
        for (int j = 0; j < 4; ++j) {
            a[i][j]     = rt[i * 4 + j];
            a[i][4 + j] = (i == j) ? 1.0f : 0.0f;
        }
    for (int c = 0; c < 4; ++c) {
        int piv = c;
        for (int r = c + 1; r < 4; ++r)
            if (fabsf(a[r][c]) > fabsf(a[piv][c])) piv = r;
        if (piv != c)
            for (int j = 0; j < 8; ++j) { float t = a[c][j]; a[c][j] = a[piv][j]; a[piv][j] = t; }
        float inv = 1.0f / a[c][c];
        for (int j = 0; j < 8; ++j) a[c][j] *= inv;
        for (int r = 0; r < 4; ++r) {
            if (r == c) continue;
            float f = a[r][c];
            for (int j = 0; j < 8; ++j) a[r][j] -= f * a[c][j];
        }
    }
    // c = (inv(rt) @ [iso,1])[:3]
    float cp[3];
    for (int i = 0; i < 3; ++i)
        cp[i] = a[i][4] * iso[0] + a[i][5] * iso[1] + a[i][6] * iso[2] + a[i][7];

    // ---- so3_exp ----
    float wx = rot[0], wy = rot[1], wz = rot[2];
    float th2 = wx * wx + wy * wy + wz * wz;
    float th  = sqrtf(th2);
    float A, B;
    if (th < 1e-6f) {
        A = 1.0f - th2 / 6.0f;
        B = 0.5f - th2 / 24.0f;
    } else {
        A = sinf(th) / th;
        B = (1.0f - cosf(th)) / th2;
    }
    float K[3][3]  = {{0.f, -wz, wy}, {wz, 0.f, -wx}, {-wy, wx, 0.f}};
    float K2[3][3];
    for (int i = 0; i < 3; ++i)
        for (int j = 0; j < 3; ++j)
            K2[i][j] = K[i][0] * K[0][j] + K[i][1] * K[1][j] + K[i][2] * K[2][j];
    float R[3][3];
    for (int i = 0; i < 3; ++i)
        for (int j = 0; j < 3; ++j)
            R[i][j] = ((i == j) ? 1.0f : 0.0f) + A * K[i][j] + B * K2[i][j];

    // t = R @ xyz ; pose translation pt = c + t - R@c
    float pt[3];
    for (int i = 0; i < 3; ++i) {
        float t_i  = R[i][0] * xyz[0] + R[i][1] * xyz[1] + R[i][2] * xyz[2];
        float Rc_i = R[i][0] * cp[0] + R[i][1] * cp[1] + R[i][2] * cp[2];
        pt[i] = cp[i] + t_i - Rc_i;
    }
    // pose P = [[R, pt],[0,0,0,1]];  ext E = rt @ P
    float P[4][4] = {{R[0][0], R[0][1], R[0][2], pt[0]},
                     {R[1][0], R[1][1], R[1][2], pt[1]},
                     {R[2][0], R[2][1], R[2][2], pt[2]},
                     {0.f, 0.f, 0.f, 1.f}};
    float E[4][4];
    for (int i = 0; i < 4; ++i)
        for (int j = 0; j < 4; ++j) {
            float s = 0.f;
            for (int k = 0; k < 4; ++k) s += rt[i * 4 + k] * P[k][j];
            E[i][j] = s;
        }

    float sdd = sddp[0];
    // M3 = sdd * E[:3,:3] @ kinv
    for (int i = 0; i < 3; ++i)
        for (int j = 0; j < 3; ++j) {
            float s = E[i][0] * kinv[0 * 3 + j] + E[i][1] * kinv[1 * 3 + j] + E[i][2] * kinv[2 * 3 + j];
            ws[i * 3 + j] = sdd * s;
        }
    ws[9]  = E[0][3];
    ws[10] = E[1][3];
    ws[11] = E[2][3];
}

// ---------------------------------------------------------------------------
// Trilinear sampling, map_coordinates(order=1, mode='constant', cval=0)
// ---------------------------------------------------------------------------
__device__ __forceinline__ float tri_sample(const float* __restrict__ vol,
                                            float x, float y, float z) {
    float fx = floorf(x), fy = floorf(y), fz = floorf(z);
    int ix = (int)fx, iy = (int)fy, iz = (int)fz;
    float wx = x - fx, wy = y - fy, wz = z - fz;

    if (ix >= 0 && ix < VD - 1 && iy >= 0 && iy < VD - 1 && iz >= 0 && iz < VD - 1) {
        const float* p = vol + ((ix * VD + iy) * VD + iz);
        float v000 = p[0],            v001 = p[1];
        float v010 = p[VD],           v011 = p[VD + 1];
        float v100 = p[VD * VD],      v101 = p[VD * VD + 1];
        float v110 = p[VD * VD + VD], v111 = p[VD * VD + VD + 1];
        float c00 = fmaf(wz, v001 - v000, v000);
        float c01 = fmaf(wz, v011 - v010, v010);
        float c10 = fmaf(wz, v101 - v100, v100);
        float c11 = fmaf(wz, v111 - v110, v110);
        float c0  = fmaf(wy, c01 - c00, c00);
        float c1  = fmaf(wy, c11 - c10, c10);
        return fmaf(wx, c1 - c0, c0);
    }
    // fully-outside reject (all corner weights land on cval=0)
    if (x <= -1.f || y <= -1.f || z <= -1.f ||
        x >= (float)VD || y >= (float)VD || z >= (float)VD) return 0.f;

    auto at = [&](int i, int j, int k) -> float {
        if ((unsigned)i < VD && (unsigned)j < VD && (unsigned)k < VD)
            return vol[(i * VD + j) * VD + k];
        return 0.f;
    };
    float v000 = at(ix, iy, iz),         v001 = at(ix, iy, iz + 1);
    float v010 = at(ix, iy + 1, iz),     v011 = at(ix, iy + 1, iz + 1);
    float v100 = at(ix + 1, iy, iz),     v101 = at(ix + 1, iy, iz + 1);
    float v110 = at(ix + 1, iy + 1, iz), v111 = at(ix + 1, iy + 1, iz + 1);
    float c00 = fmaf(wz, v001 - v000, v000);
    float c01 = fmaf(wz, v011 - v010, v010);
    float c10 = fmaf(wz, v101 - v100, v100);
    float c11 = fmaf(wz, v111 - v110, v110);
    float c0  = fmaf(wy, c01 - c00, c00);
    float c1  = fmaf(wy, c11 - c10, c10);
    return fmaf(wx, c1 - c0, c0);
}

// ---------------------------------------------------------------------------
// Kernel 2: ray march. One wave handles a 16x2 pixel strip; seg = M3 @ [u,v,1]
// for all 32 rays is computed with two V_WMMA_F32_16X16X4_F32 ops.
// ---------------------------------------------------------------------------
__global__ __launch_bounds__(256)
void reg_render_kernel(const float* __restrict__ vol,
                       const float* __restrict__ ws,
                       const int* __restrict__ Hp, const int* __restrict__ Wp,
                       const int* __restrict__ Np,
                       float* __restrict__ out) {
    const int H = Hp[0], W = Wp[0], N = Np[0];

    // uniform ray-setup constants
    const float m00 = ws[0], m01 = ws[1], m02 = ws[2];
    const float m10 = ws[3], m11 = ws[4], m12 = ws[5];
    const float m20 = ws[6], m21 = ws[7], m22 = ws[8];
    const float srcx = ws[9], srcy = ws[10], srcz = ws[11];

    const int lane = threadIdx.x & 31;
    const int wv   = threadIdx.x >> 5;      // wave id in block: rows 2wv, 2wv+1
    const int half = lane >> 4;
    const int li   = lane & 15;

    const int tileX = blockIdx.x * TILE;
    const int tileY = blockIdx.y * TILE;

    // ---- A matrix (16x4): A[m][k] = M3[m][k] for m,k < 3 else 0 ----
    // layout: lanes 0-15 hold K=0 (v0), K=1 (v1); lanes 16-31 hold K=2,K=3
    float ax_r0 = half ? m02 : m00, ax_r1 = half ? m12 : m10, ax_r2 = half ? m22 : m20;
    float ay_r0 = half ? 0.f : m01, ay_r1 = half ? 0.f : m11, ay_r2 = half ? 0.f : m21;
    v2f a;
    a.x = (li == 0) ? ax_r0 : (li == 1) ? ax_r1 : (li == 2) ? ax_r2 : 0.f;
    a.y = (li == 0) ? ay_r0 : (li == 1) ? ay_r1 : (li == 2) ? ay_r2 : 0.f;

    // ---- B matrices (4x16): column n = [u_n, v_n, 1, 0] ----
    // layout: v0 = K=0 (lanes 0-15) / K=2 (lanes 16-31); v1 = K=1 / K=3
    const float u  = (float)(tileX + li) + 0.5f;
    const float v0 = (float)(tileY + 2 * wv) + 0.5f;
    v2f b0, b1;
    b0.x = half ? 1.0f : u;
    b0.y = half ? 0.0f : v0;
    b1.x = half ? 1.0f : u;
    b1.y = half ? 0.0f : (v0 + 1.0f);

    v8f d0 = {}, d1 = {};
    d0 = __builtin_amdgcn_wmma_f32_16x16x4_f32(false, a, false, b0, (short)0, d0, false, false);
    d1 = __builtin_amdgcn_wmma_f32_16x16x4_f32(false, a, false, b1, (short)0, d1, false, false);

    // distribute: lane L wants pixel (col=L&15, row=2wv + (L>>4))
    float sx0 = __shfl(d0[0], li, 32), sx1 = __shfl(d1[0], li, 32);
    float sy0 = __shfl(d0[1], li, 32), sy1 = __shfl(d1[1], li, 32);
    float sz0 = __shfl(d0[2], li, 32), sz1 = __shfl(d1[2], li, 32);
    const float segx = half ? sx1 : sx0;
    const float segy = half ? sy1 : sy0;
    const float segz = half ? sz1 : sz0;

    const int col = tileX + li;
    const int row = tileY + 2 * wv + half;
    if (col >= W || row >= H) return;   // EXEC was full through the WMMAs

    const float invN = 1.0f / (float)N;
    const float step = sqrtf(segx * segx + segy * segy + segz * segz) * invN;

    float acc = 0.f;
    int s = 0;
    for (; s + 4 <= N; s += 4) {
        // prefetch the base voxel 8 samples ahead along the ray
        {
            float tp = ((float)s + 8.5f) * invN;
            float xp = fmaf(tp, segx, srcx);
            float yp = fmaf(tp, segy, srcy);
            float zp = fmaf(tp, segz, srcz);
            int ixp = (int)floorf(xp), iyp = (int)floorf(yp), izp = (int)floorf(zp);
            if ((unsigned)ixp < VD && (unsigned)iyp < VD && (unsigned)izp < VD)
                __builtin_prefetch(vol + ((ixp * VD + iyp) * VD + izp), 0, 3);
        }
#pragma unroll
        for (int j = 0; j < 4; ++j) {
            float t = ((float)(s + j) + 0.5f) * invN;
            float x = fmaf(t, segx, srcx);
            float y = fmaf(t, segy, srcy);
            float z = fmaf(t, segz, srcz);
            acc += tri_sample(vol, x, y, z);
        }
    }
    for (; s < N; ++s) {
        float t = ((float)s + 0.5f) * invN;
        acc += tri_sample(vol, fmaf(t, segx, srcx), fmaf(t, segy, srcy), fmaf(t, segz, srcz));
    }

    out[row * W + col] = acc * step;
}

// ---------------------------------------------------------------------------
extern "C" void kernel_launch(void* const* d_in, const int* in_sizes, int n_in,
                              void* d_out, int out_size, void* d_ws, size_t ws_size,
                              hipStream_t stream) {
    const float* vol  = (const float*)d_in[0];
    const float* iso  = (const float*)d_in[1];
    const float* rt   = (const float*)d_in[2];
    const float* kinv = (const float*)d_in[3];
    const float* sdd  = (const float*)d_in[4];
    const float* rot  = (const float*)d_in[5];
    const float* xyz  = (const float*)d_in[6];
    const int*   Hp   = (const int*)d_in[7];
    const int*   Wp   = (const int*)d_in[8];
    const int*   Np   = (const int*)d_in[9];
    float* out = (float*)d_out;
    float* ws  = (float*)d_ws;

    reg_setup_kernel<<<1, 32, 0, stream>>>(iso, rt, kinv, sdd, rot, xyz, ws);

    // output is H x W with H == W (512x512 for this problem)
    int Wd = (int)(sqrt((double)out_size) + 0.5);
    int Hd = out_size / Wd;
    dim3 grid((Wd + TILE - 1) / TILE, (Hd + TILE - 1) / TILE);
    reg_render_kernel<<<grid, 256, 0, stream>>>(vol, ws, Hp, Wp, Np, out);
}